// ViterbiLoss_9285719294228
// MI455X (gfx1250) — compile-verified
//
#include <hip/hip_runtime.h>

// ViterbiLoss (linear-chain CRF forward + gold score), MI455X / gfx1250.
// B=256, T=512, K=20. Memory-bound: 210MB of features @ 23.3TB/s ~ 9us floor.
// Strategy: associative scaled matrix-product forward algorithm.
//   exp(alpha_t) = P_t * exp(alpha_{t-1}),  P_t = exp(features[b,t,:,:])
// Split T into NC chunks -> B*NC = 4096 independent waves, each computing the
// rescaled 32x32 (zero-padded from 20x20) chunk product with fp32 WMMA
// (V_WMMA_F32_16X16X4_F32). Feature staging uses double-buffered
// GLOBAL_LOAD_ASYNC_TO_LDS_B128 (ASYNCcnt DMA) so the next timestep's 1600B
// block streams into LDS while the current step's WMMAs run.

#define Bn 256
#define Tn 512
#define Kn 20
#define START_TAG 18
#define STOP_TAG 19
#define NC 16
#define SPC 32              // NC * SPC == Tn
#define NEG_INF (-__builtin_inff())

typedef __attribute__((ext_vector_type(2))) float v2f;
typedef __attribute__((ext_vector_type(8))) float v8f;

__device__ __forceinline__ v8f wmma_f32_16x16x4(v2f a, v2f b, v8f c) {
    // 8 args: (neg_a, A, neg_b, B, c_mod, C, reuse_a, reuse_b)
    return __builtin_amdgcn_wmma_f32_16x16x4_f32(false, a, false, b,
                                                 (short)0, c, false, false);
}

// Async DMA: 16B from global (per active lane) into LDS at per-lane address.
__device__ __forceinline__ void async_ld_b128(unsigned lds_addr, const void* g) {
    asm volatile("global_load_async_to_lds_b128 %0, %1, off"
                 :: "v"(lds_addr), "v"(g) : "memory");
}

__device__ __forceinline__ float wave_max(float x) {
#pragma unroll
    for (int m = 16; m; m >>= 1) x = fmaxf(x, __shfl_xor(x, m, 32));
    return x;
}
__device__ __forceinline__ float wave_sum(float x) {
#pragma unroll
    for (int m = 16; m; m >>= 1) x += __shfl_xor(x, m, 32);
    return x;
}

// ---------------------------------------------------------------------------
// Kernel 1: per-(b, chunk) rescaled product of exp(feature) matrices.
// One wave per block. LDS: double-buffered staging tile (raw feats, -inf
// padded, filled by async DMA) + relayout tile (column-major so B-fragment
// pairs are b64-contiguous). Running product M held as WMMA B fragments:
//   lane<16 : VGPR g holds M[4kc+g  ][nt*16 + lane]
//   lane>=16: VGPR g holds M[4kc+2+g][nt*16 + lane-16]
// A operand (P) per documented 32-bit 16x4 layout (same koff symmetry).
// ---------------------------------------------------------------------------
__global__ __launch_bounds__(32)
void crf_chunk_products(const float* __restrict__ feats,
                        const int* __restrict__ lengths,
                        float* __restrict__ wsR,
                        float* __restrict__ wsS) {
    __shared__ float Pt[2][32 * 32]; // raw features, padding = -inf (exp -> 0)
    __shared__ float Rt[32 * 32];    // column-major: Rt[n*32 + k] = M[k][n]
    const int l = threadIdx.x;
    const int bc = blockIdx.x;
    const int b = bc / NC;
    const int c = bc % NC;
    const int len = lengths[b];
    int t0 = c * SPC; if (t0 < 1) t0 = 1;           // t=0 feeds alpha0 only
    int t1 = (c + 1) * SPC; if (t1 > len) t1 = len; // masked steps are identity

    const int col  = l & 15;        // within-tile column / A-row
    const int koff = (l >> 4) * 2;  // K sub-offset per lane half

    // Per-lane async-copy addressing: the 20x20 block is 1600 contiguous
    // bytes; lanes 0..24 each move 64B as 4x16B pieces. 80B source rows are
    // 16B-aligned, so each piece maps to one LDS row (stride 128B, padded).
    unsigned ldsA[2][4];
    int goff[4];
    {
        const unsigned ptBase = (unsigned)(uintptr_t)(&Pt[0][0]);
#pragma unroll
        for (int q = 0; q < 4; ++q) {
            const int p = l * 64 + q * 16;     // byte offset into 1600B block
            goff[q] = p;
            const int row = p / 80;
            const int jb  = p - row * 80;
            ldsA[0][q] = ptBase + row * 128 + jb;
            ldsA[1][q] = ptBase + 4096 + row * 128 + jb;
        }
    }

    // M = zero-padded 20x20 identity, as B fragments.
    v2f Rf[8][2];
#pragma unroll
    for (int kc = 0; kc < 8; ++kc)
#pragma unroll
        for (int nt = 0; nt < 2; ++nt) {
            int k = kc * 4 + koff;
            int n = nt * 16 + col;
            Rf[kc][nt].x = (k     == n && n < Kn) ? 1.0f : 0.0f;
            Rf[kc][nt].y = (k + 1 == n && n < Kn) ? 1.0f : 0.0f;
        }
    float logscale = 0.0f;

    // Pad only the region the DMA never writes (rows>=20 or cols>=20), both
    // buffers, so padding stores and async DMA touch disjoint LDS bytes.
    {
        float* pt = &Pt[0][0];
        for (int e = l; e < 2048; e += 32) {
            const int i = (e >> 5) & 31, j = e & 31;
            if (i >= Kn || j >= Kn) pt[e] = NEG_INF;
        }
    }
    // Kick off DMA for the first timestep.
    if (t0 < t1) {
        const char* g = (const char*)(feats + (size_t)(b * Tn + t0) * (Kn * Kn));
        if (l < 25) {
            async_ld_b128(ldsA[t0 & 1][0], g + goff[0]);
            async_ld_b128(ldsA[t0 & 1][1], g + goff[1]);
            async_ld_b128(ldsA[t0 & 1][2], g + goff[2]);
            async_ld_b128(ldsA[t0 & 1][3], g + goff[3]);
        }
    }
    __syncthreads();

    for (int t = t0; t < t1; ++t) {
        const int cb = t & 1;
        // Prefetch next timestep into the other buffer, then retire only the
        // current buffer's 4 DMAs (async loads complete in order).
        if (t + 1 < t1) {
            const char* g = (const char*)(feats + (size_t)(b * Tn + t + 1) * (Kn * Kn));
            if (l < 25) {
                async_ld_b128(ldsA[cb ^ 1][0], g + goff[0]);
                async_ld_b128(ldsA[cb ^ 1][1], g + goff[1]);
                async_ld_b128(ldsA[cb ^ 1][2], g + goff[2]);
                async_ld_b128(ldsA[cb ^ 1][3], g + goff[3]);
            }
            asm volatile("s_wait_asynccnt 0x4" ::: "memory");
        } else {
            asm volatile("s_wait_asynccnt 0x0" ::: "memory");
        }

        const float* P = &Pt[cb][0];

        // D = exp(P) * M  (2x2 output tiles, K swept in chunks of 4)
        v8f d[2][2] = {};
#pragma unroll
        for (int kc = 0; kc < 8; ++kc) {
            const int kk = kc * 4 + koff;
            v2f a0, a1;
            a0.x = __expf(P[col * 32 + kk]);
            a0.y = __expf(P[col * 32 + kk + 1]);
            a1.x = __expf(P[(16 + col) * 32 + kk]);
            a1.y = __expf(P[(16 + col) * 32 + kk + 1]);
            d[0][0] = wmma_f32_16x16x4(a0, Rf[kc][0], d[0][0]);
            d[0][1] = wmma_f32_16x16x4(a0, Rf[kc][1], d[0][1]);
            d[1][0] = wmma_f32_16x16x4(a1, Rf[kc][0], d[1][0]);
            d[1][1] = wmma_f32_16x16x4(a1, Rf[kc][1], d[1][1]);
        }

        // Relayout D -> B via LDS; track max for per-step rescaling.
        float mx = 0.0f;
#pragma unroll
        for (int rt = 0; rt < 2; ++rt)
#pragma unroll
            for (int nt = 0; nt < 2; ++nt)
#pragma unroll
                for (int v = 0; v < 8; ++v) {
                    float val = d[rt][nt][v];
                    mx = fmaxf(mx, val);
                    // D layout: row m = rt*16 + v + (lane>=16 ? 8 : 0)
                    Rt[(nt * 16 + col) * 32 + rt * 16 + (l >> 4) * 8 + v] = val;
                }
        mx = fmaxf(wave_max(mx), 1e-30f);
        const float inv = 1.0f / mx;
        logscale += __logf(mx);
        __syncthreads();
#pragma unroll
        for (int kc = 0; kc < 8; ++kc) {
            const int k = kc * 4 + koff;
#pragma unroll
            for (int nt = 0; nt < 2; ++nt) {
                const int n = nt * 16 + col;
                Rf[kc][nt].x = Rt[n * 32 + k]     * inv;   // b64-contiguous
                Rf[kc][nt].y = Rt[n * 32 + k + 1] * inv;
            }
        }
        __syncthreads();   // drain DS reads before next DMA overwrites Pt[cb]
    }

    // Emit M (row-major M[k][n]) + accumulated log-scale.
    float* out = wsR + (size_t)bc * 1024;
#pragma unroll
    for (int kc = 0; kc < 8; ++kc) {
        const int k = kc * 4 + koff;
#pragma unroll
        for (int nt = 0; nt < 2; ++nt) {
            const int n = nt * 16 + col;
            out[k * 32 + n]       = Rf[kc][nt].x;
            out[(k + 1) * 32 + n] = Rf[kc][nt].y;
        }
    }
    if (l == 0) wsS[bc] = logscale;
}

// ---------------------------------------------------------------------------
// Kernel 2: one wave per batch element. Fold alpha0 through the 16 chunk
// matrices (scaled matvec), finish logsumexp with STOP transitions, and
// compute the gold score. Deterministic (fixed shuffle trees, no atomics).
// ---------------------------------------------------------------------------
__global__ __launch_bounds__(32)
void crf_combine(const float* __restrict__ feats,
                 const float* __restrict__ trans,
                 const int* __restrict__ targets,
                 const int* __restrict__ lengths,
                 const float* __restrict__ wsR,
                 const float* __restrict__ wsS,
                 float* __restrict__ wsP) {
    const int b = blockIdx.x;
    const int l = threadIdx.x;
    const int len = lengths[b];

    // alpha0[j] = features[b,0,j,START]
    float a0 = (l < Kn) ? feats[((size_t)(b * Tn) * Kn + l) * Kn + START_TAG]
                        : NEG_INF;
    const float m0 = wave_max(a0);
    float v = (l < Kn) ? __expf(a0 - m0) : 0.0f;   // lane j holds exp(alpha[j]-m)
    float logacc = m0;

    for (int c = 0; c < NC; ++c) {
        const float* M   = wsR + ((size_t)b * NC + c) * 1024;
        const float* row = M + l * 32;             // lane i <- row i of M
        float w = 0.0f;
#pragma unroll 8
        for (int j = 0; j < 32; ++j)
            w += row[j] * __shfl(v, j, 32);        // rows >=20 are all-zero
        const float mx = fmaxf(wave_max(w), 1e-30f);
        v = w / mx;
        logacc += __logf(mx) + wsS[b * NC + c];
    }

    // all_paths_b = logsumexp_i( log v_i + logacc + trans[STOP,i] )
    float al = (l < Kn) ? (__logf(fmaxf(v, 1e-37f)) + logacc + trans[STOP_TAG * Kn + l])
                        : NEG_INF;
    const float zm = wave_max(al);
    const float se = wave_sum((l < Kn) ? __expf(al - zm) : 0.0f);
    const float all_paths = zm + __logf(se);

    // gold_b = sum_{t<len} features[b,t,tgt,prev] + trans[STOP,last]
    float g = 0.0f;
    for (int t = l; t < len; t += 32) {
        const int tgt  = targets[b * Tn + t];
        const int prev = (t == 0) ? START_TAG : targets[b * Tn + t - 1];
        g += feats[((size_t)(b * Tn + t) * Kn + tgt) * Kn + prev];
    }
    g = wave_sum(g);

    if (l == 0) {
        const int last = targets[b * Tn + len - 1];
        g += trans[STOP_TAG * Kn + last];
        wsP[b] = all_paths - g;
    }
}

// ---------------------------------------------------------------------------
// Kernel 3: deterministic fixed-order reduction of the 256 per-b partials.
// ---------------------------------------------------------------------------
__global__ __launch_bounds__(32)
void crf_final(const float* __restrict__ wsP, float* __restrict__ out) {
    const int l = threadIdx.x;
    float s = 0.0f;
    for (int i = l; i < Bn; i += 32) s += wsP[i];
    s = wave_sum(s);
    if (l == 0) out[0] = s;
}

// ---------------------------------------------------------------------------
extern "C" void kernel_launch(void* const* d_in, const int* in_sizes, int n_in,
                              void* d_out, int out_size, void* d_ws, size_t ws_size,
                              hipStream_t stream) {
    (void)in_sizes; (void)n_in; (void)out_size; (void)ws_size;
    const float* feats   = (const float*)d_in[0];
    const float* trans   = (const float*)d_in[1];
    const int*   targets = (const int*)d_in[2];
    const int*   lengths = (const int*)d_in[3];
    float* out = (float*)d_out;

    // Workspace: chunk matrices (16.78 MB) + scales + per-b partials.
    float* wsR = (float*)d_ws;
    float* wsS = wsR + (size_t)Bn * NC * 1024;
    float* wsP = wsS + (size_t)Bn * NC;

    crf_chunk_products<<<Bn * NC, 32, 0, stream>>>(feats, lengths, wsR, wsS);
    crf_combine<<<Bn, 32, 0, stream>>>(feats, trans, targets, lengths, wsR, wsS, wsP);
    crf_final<<<1, 32, 0, stream>>>(wsP, out);
}